// AMS_69243462746520
// MI455X (gfx1250) — compile-verified
//
#include <hip/hip_runtime.h>
#include <math.h>

typedef __attribute__((ext_vector_type(2))) float v2f;
typedef __attribute__((ext_vector_type(8))) float v8f;

#define BB 512
#define LL 96
#define NN 32
#define CC 16
#define EE 4

// ---------------------------------------------------------------------------
// Kernel 0: zero the importance/load accumulators (must run every call).
// ---------------------------------------------------------------------------
__global__ void ams_init_kernel(float* imp, float* loadv) {
    int t = threadIdx.x;
    if (t < EE) { imp[t] = 0.0f; loadv[t] = 0.0f; }
}

// ---------------------------------------------------------------------------
// Kernel 1: gather xs = x[...,0] into (B, N, L) layout. Also warms L2 with x.
// ---------------------------------------------------------------------------
__global__ void ams_xs_gather_kernel(const float* __restrict__ x,
                                     float* __restrict__ xs_t, int total) {
    int i = blockIdx.x * blockDim.x + threadIdx.x;
    if (i >= total) return;
    int n = i & (NN - 1);            // x linear index = ((b*L + l)*N + n)
    int l = (i >> 5) % LL;
    int b = i / (LL * NN);
    xs_t[((size_t)b * NN + n) * LL + l] = x[(size_t)i * CC];
}

// ---------------------------------------------------------------------------
// Kernel 2: per-batch gating pipeline.
//   trend (moving avgs k=4,8,12, edge pad) + 96-pt DFT top-3 seasonality
//   -> g[b,l] -> logits -> top-2 softmax gates -> W_eff[b] (16x16, +I),
//   b_eff[b] (16), atomic importance/load accumulation.
// One block of 96 threads per batch row b; thread index == l.
// ---------------------------------------------------------------------------
__global__ __launch_bounds__(LL) void ams_gate_kernel(
    const float* __restrict__ xs_t, const float* __restrict__ W_start,
    const float* __restrict__ b_start, const float* __restrict__ W_gate,
    const float* __restrict__ b_gate, const float* __restrict__ W_exp,
    const float* __restrict__ b_exp, float* __restrict__ Weff,
    float* __restrict__ beff, float* imp, float* loadv) {
    __shared__ float s_x[LL];
    __shared__ float s_re[47], s_im[47], s_mag[47];
    __shared__ float s_g[LL];
    __shared__ float s_logit[EE];
    __shared__ float s_gates[EE];
    __shared__ int   s_top[3];

    const int b = blockIdx.x;
    const int l = threadIdx.x;
    const float PI2 = 6.28318530717958647692f;

    float gacc = 0.0f;
    for (int n = 0; n < NN; ++n) {
        s_x[l] = xs_t[((size_t)b * NN + n) * LL + l];
        __syncthreads();

        // trend: mean of 3 edge-padded moving averages
        float tr = 0.0f;
        const int ks[3] = {4, 8, 12};
        #pragma unroll
        for (int j = 0; j < 3; ++j) {
            int k = ks[j];
            int pf = (k - 1) / 2, pe = k / 2;
            float s = 0.0f;
            for (int d = -pf; d <= pe; ++d) {
                int p = l + d;
                p = p < 0 ? 0 : (p > LL - 1 ? LL - 1 : p);
                s += s_x[p];
            }
            tr += s / (float)k;
        }
        tr *= (1.0f / 3.0f);

        // DFT bins 1..47 (drop DC and Nyquist for even L)
        if (l < 47) {
            int k = l + 1;
            float re = 0.0f, im = 0.0f;
            float w = -PI2 * (float)k / (float)LL;
            for (int t = 0; t < LL; ++t) {
                float sc, cc2;
                __sincosf(w * (float)t, &sc, &cc2);
                re += s_x[t] * cc2;
                im += s_x[t] * sc;
            }
            s_re[l] = re; s_im[l] = im;
            s_mag[l] = sqrtf(re * re + im * im);
        }
        __syncthreads();

        // top-3 magnitudes (first occurrence wins, matching top_k)
        if (l == 0) {
            for (int j = 0; j < 3; ++j) {
                float mb = -1.0f; int ib = 0;
                for (int k2 = 0; k2 < 47; ++k2) {
                    bool used = false;
                    for (int q = 0; q < j; ++q) used |= (s_top[q] == k2);
                    if (!used && s_mag[k2] > mb) { mb = s_mag[k2]; ib = k2; }
                }
                s_top[j] = ib;
            }
        }
        __syncthreads();

        // seasonality reconstruction
        float wave = 0.0f;
        #pragma unroll
        for (int j = 0; j < 3; ++j) {
            int kk = s_top[j];
            float amp = s_mag[kk] * (1.0f / (float)LL);
            float ph = atan2f(s_im[kk], s_re[kk]);
            float f = (float)(kk + 1) / (float)LL;
            wave += 2.0f * amp * __cosf(PI2 * f * (float)l + ph);
        }

        gacc += (s_x[l] + wave + tr) * W_start[n];
        __syncthreads();   // protect s_x / s_re / s_top before next n
    }

    s_g[l] = gacc + b_start[0];
    __syncthreads();

    if (l < EE) {
        float acc = b_gate[l];
        for (int t = 0; t < LL; ++t) acc += s_g[t] * W_gate[l * LL + t];
        s_logit[l] = acc;
    }
    __syncthreads();

    if (l == 0) {
        int i0 = 0;
        for (int e = 1; e < EE; ++e) if (s_logit[e] > s_logit[i0]) i0 = e;
        int i1 = -1;
        for (int e = 0; e < EE; ++e) {
            if (e == i0) continue;
            if (i1 < 0 || s_logit[e] > s_logit[i1]) i1 = e;
        }
        float e1 = __expf(s_logit[i1] - s_logit[i0]);
        float g0 = 1.0f / (1.0f + e1);
        float g1 = e1 / (1.0f + e1);
        for (int e = 0; e < EE; ++e) s_gates[e] = 0.0f;
        s_gates[i0] = g0; s_gates[i1] = g1;
        for (int e = 0; e < EE; ++e) {
            atomicAdd(&imp[e], s_gates[e]);
            atomicAdd(&loadv[e], s_gates[e] > 0.0f ? 1.0f : 0.0f);
        }
    }
    __syncthreads();

    // W_eff[b] = sum_e gate_e * W_exp[e] + I ;  b_eff[b] = sum_e gate_e * b_exp[e]
    for (int idx = l; idx < CC * CC; idx += LL) {
        int c = idx >> 4, f = idx & 15;
        float w = (c == f) ? 1.0f : 0.0f;
        for (int e = 0; e < EE; ++e) w += s_gates[e] * W_exp[e * CC * CC + idx];
        Weff[(size_t)b * CC * CC + idx] = w;
    }
    if (l < CC) {
        float w = 0.0f;
        for (int e = 0; e < EE; ++e) w += s_gates[e] * b_exp[e * CC + l];
        beff[(size_t)b * CC + l] = w;
    }
}

// ---------------------------------------------------------------------------
// Kernel 3: out[b,r,:] = x[b,r,:] @ W_eff[b] + b_eff[b]   via WMMA f32 16x16x4.
// Grid (B, 4): each block (8 waves) handles 48 of the 192 16-row tiles of one
// batch; 6 tiles per wave. x loads are non-temporal (last use), out stores are
// non-temporal so they evict first from L2 and keep x resident (L2 = 192 MB,
// x = 100 MB pre-warmed by the gather kernel).
// ---------------------------------------------------------------------------
__global__ __launch_bounds__(256) void ams_moe_wmma_kernel(
    const float* __restrict__ x, const float* __restrict__ Weff,
    const float* __restrict__ beff, float* __restrict__ out) {
    const int b = blockIdx.x;
    const int wave = __builtin_amdgcn_readfirstlane(threadIdx.x >> 5); // uniform
    const int lane = threadIdx.x & 31;
    const int lo = lane & 15;   // col within tile (B/C/D), row within A
    const int hi = lane >> 4;   // half-wave select

    // B fragments: 4x16 chunk kk; VGPR0 -> rows {4k+0, 4k+2}, VGPR1 -> {4k+1, 4k+3}
    const float* W = Weff + (size_t)b * CC * CC;
    v2f bf[4];
    #pragma unroll
    for (int kk = 0; kk < 4; ++kk) {
        int r0 = kk * 4 + 2 * hi;
        bf[kk].x = W[(r0 + 0) * CC + lo];
        bf[kk].y = W[(r0 + 1) * CC + lo];
    }
    const float bias = beff[(size_t)b * CC + lo];

    const int t_end = 48 * (blockIdx.y + 1);
    for (int t = wave + 48 * blockIdx.y; t < t_end; t += 8) {
        size_t base = ((size_t)b * (LL * NN) + (size_t)t * 16) * CC;

        v8f c;
        #pragma unroll
        for (int r = 0; r < 8; ++r) c[r] = bias;

        #pragma unroll
        for (int kk = 0; kk < 4; ++kk) {
            // A 16x4 fp32: lane holds row lo, K = 4*kk + 2*hi + {0,1}
            const v2f* ap = (const v2f*)(x + base + (size_t)lo * CC + kk * 4 + 2 * hi);
            v2f a = __builtin_nontemporal_load(ap);
            c = __builtin_amdgcn_wmma_f32_16x16x4_f32(
                    false, a, false, bf[kk], (short)0, c, false, false);
        }

        // D: VGPR r -> row r + 8*hi, col lo
        size_t obase = base + (size_t)(8 * hi) * CC + lo;
        #pragma unroll
        for (int r = 0; r < 8; ++r)
            __builtin_nontemporal_store(c[r], &out[obase + (size_t)r * CC]);
    }
}

// ---------------------------------------------------------------------------
// Kernel 4: balance loss = 0.01 * (cv2(importance) + cv2(load)), ddof=1.
// ---------------------------------------------------------------------------
__global__ void ams_loss_kernel(const float* imp, const float* loadv,
                                float* out_loss) {
    if (threadIdx.x == 0) {
        float mi = 0.0f, ml = 0.0f;
        for (int e = 0; e < EE; ++e) { mi += imp[e]; ml += loadv[e]; }
        mi *= (1.0f / EE); ml *= (1.0f / EE);
        float vi = 0.0f, vl = 0.0f;
        for (int e = 0; e < EE; ++e) {
            float d = imp[e] - mi;  vi += d * d;
            d = loadv[e] - ml;      vl += d * d;
        }
        vi /= (float)(EE - 1); vl /= (float)(EE - 1);
        *out_loss = 0.01f * (vi / (mi * mi + 1e-10f) + vl / (ml * ml + 1e-10f));
    }
}

extern "C" void kernel_launch(void* const* d_in, const int* in_sizes, int n_in,
                              void* d_out, int out_size, void* d_ws, size_t ws_size,
                              hipStream_t stream) {
    (void)in_sizes; (void)n_in; (void)out_size; (void)ws_size;
    const float* x       = (const float*)d_in[0];
    const float* W_start = (const float*)d_in[1];
    const float* b_start = (const float*)d_in[2];
    const float* W_gate  = (const float*)d_in[3];
    const float* b_gate  = (const float*)d_in[4];
    const float* W_exp   = (const float*)d_in[5];
    const float* b_exp   = (const float*)d_in[6];
    float* out = (float*)d_out;

    // workspace layout
    float* ws    = (float*)d_ws;
    float* xs_t  = ws;                                  // B*N*L
    float* Weff  = xs_t + (size_t)BB * NN * LL;         // B*256
    float* beff  = Weff + (size_t)BB * CC * CC;         // B*16
    float* imp   = beff + (size_t)BB * CC;              // 4
    float* loadv = imp + EE;                            // 4

    ams_init_kernel<<<1, 32, 0, stream>>>(imp, loadv);

    int total = BB * LL * NN;
    ams_xs_gather_kernel<<<(total + 255) / 256, 256, 0, stream>>>(x, xs_t, total);

    ams_gate_kernel<<<BB, LL, 0, stream>>>(xs_t, W_start, b_start, W_gate,
                                           b_gate, W_exp, b_exp, Weff, beff,
                                           imp, loadv);

    ams_moe_wmma_kernel<<<dim3(BB, 4), 256, 0, stream>>>(x, Weff, beff, out);

    ams_loss_kernel<<<1, 32, 0, stream>>>(imp, loadv,
                                          out + (size_t)BB * LL * NN * CC);
}